// RoPEAttention3D_824633720905
// MI455X (gfx1250) — compile-verified
//
#include <hip/hip_runtime.h>
#include <hip/hip_bf16.h>
#include <math.h>

// Problem constants (match reference)
constexpr int B_   = 2;
constexpr int N_   = 2048;
constexpr int DIM  = 768;
constexpr int H_   = 12;
constexpr int DH   = 64;
constexpr int LDQKV = 3 * DIM;       // 2304, row stride of qkv workspace
constexpr float EPS = 1e-6f;

typedef float v2f __attribute__((ext_vector_type(2)));
typedef float v8f __attribute__((ext_vector_type(8)));
typedef unsigned int u32x4 __attribute__((ext_vector_type(4)));
typedef int i32x4 __attribute__((ext_vector_type(4)));
typedef int i32x8 __attribute__((ext_vector_type(8)));

// D = A(16x4, f32) * B(4x16, f32) + C(16x16, f32) on the matrix pipe.
__device__ inline v8f wmma4(v2f a, v2f b, v8f c) {
    return __builtin_amdgcn_wmma_f32_16x16x4_f32(
        /*neg_a=*/false, a, /*neg_b=*/false, b,
        /*c_mod=*/(short)0, c, /*reuse_a=*/false, /*reuse_b=*/false);
}

// ---------------------------------------------------------------------------
// TDM: async-load a 16-row x 64-col fp32 tile (row stride LDQKV elements in
// global) into LDS, padded +2 DWORDs per 64 DWORDs -> LDS row stride 66 floats.
// Issued by one wave; tracked with TENSORcnt.
// D# layout per CDNA5 ISA ch.8 (group0: control/lds/global addr, group1: dims).
// ---------------------------------------------------------------------------
__device__ inline void tdm_load_tile16x64(unsigned lds_off, const float* gsrc) {
    unsigned long long ga = (unsigned long long)(uintptr_t)gsrc;
    u32x4 g0;
    g0.x = 1u;                                        // count=1: valid user D#
    g0.y = lds_off;                                   // lds_addr (bytes)
    g0.z = (unsigned)(ga & 0xffffffffu);              // global_addr[31:0]
    g0.w = (unsigned)((ga >> 32) & 0x01ffffffu)       // global_addr[56:32]
         | 0x80000000u;                               // type=2 ("image")
    i32x8 g1;
    // data_size=2 (4B) | pad_enable | pad_interval=5 (64 dw) | pad_amount=1 (2 dw)
    g1[0] = (2 << 16) | (1 << 20) | (5 << 22) | (1 << 25);
    const unsigned td0 = 64u;                         // tensor_dim0 (= tile width)
    const unsigned td1 = 1u << 20;                    // tensor_dim1: large, no OOB
    g1[1] = (int)((td0 & 0xffffu) << 16);             // [63:48]  tensor_dim0 lo
    g1[2] = (int)(((td0 >> 16) & 0xffffu) | ((td1 & 0xffffu) << 16));
    g1[3] = (int)(((td1 >> 16) & 0xffffu) | (64u << 16));   // tile_dim0 = 64
    g1[4] = 16;                                       // tile_dim1=16, tile_dim2=0
    g1[5] = LDQKV;                                    // tensor_dim0_stride = 2304
    g1[6] = 0;
    g1[7] = 0;
    i32x4 gz = {0, 0, 0, 0};
#if __clang_major__ >= 23
    i32x8 gz8 = {0, 0, 0, 0, 0, 0, 0, 0};
    __builtin_amdgcn_tensor_load_to_lds(g0, g1, gz, gz, gz8, 0);
#else
    __builtin_amdgcn_tensor_load_to_lds(g0, g1, gz, gz, 0);
#endif
    asm volatile("" ::: "memory");   // TDM writes LDS behind the compiler's back
}

// ---------------------------------------------------------------------------
// Kernel 1/4: C[M x Ncols] = A[M x K] @ W[Ncols x K]^T (+ bias)
// 8 waves/block; wave w -> rows [by*128+16w, +16), cols [bx*64, +64).
// K processed in chunks of 16 with register double-buffering so the loads for
// chunk i+1 are in flight while the 16 WMMAs of chunk i execute.
// ---------------------------------------------------------------------------
__global__ __launch_bounds__(256)
void gemm_wmma(const float* __restrict__ A, const float* __restrict__ W,
               const float* __restrict__ bias, float* __restrict__ C,
               int K, int ldc) {
    const int lane  = threadIdx.x & 31;
    const int wv    = threadIdx.x >> 5;
    const int m0    = blockIdx.y * 128 + wv * 16;
    const int n0    = blockIdx.x * 64;
    const int r16   = lane & 15;
    const int khalf = (lane >> 4) << 1;

    const float* pa = A + (size_t)(m0 + r16) * K + khalf;
    const float* pw[4];
#pragma unroll
    for (int j = 0; j < 4; ++j)
        pw[j] = W + (size_t)(n0 + 16 * j + r16) * K + khalf;

    auto ldchunk = [&](int kc, v2f a[4], v2f b[4][4]) {
#pragma unroll
        for (int t = 0; t < 4; ++t) {
            const int k0 = kc + 4 * t;
            a[t].x = pa[k0]; a[t].y = pa[k0 + 1];
#pragma unroll
            for (int j = 0; j < 4; ++j) {
                b[t][j].x = pw[j][k0]; b[t][j].y = pw[j][k0 + 1];
            }
        }
    };

    v8f acc[4] = {{}, {}, {}, {}};
    v2f a_cur[4], b_cur[4][4];
    ldchunk(0, a_cur, b_cur);

    for (int kc = 0; kc < K; kc += 16) {
        v2f a_nxt[4], b_nxt[4][4];
        const int kn = (kc + 16 < K) ? kc + 16 : kc;   // tail: re-read (unused)
        ldchunk(kn, a_nxt, b_nxt);
#pragma unroll
        for (int t = 0; t < 4; ++t)
#pragma unroll
            for (int j = 0; j < 4; ++j)
                acc[j] = wmma4(a_cur[t], b_cur[t][j], acc[j]);
#pragma unroll
        for (int t = 0; t < 4; ++t) {
            a_cur[t] = a_nxt[t];
#pragma unroll
            for (int j = 0; j < 4; ++j) b_cur[t][j] = b_nxt[t][j];
        }
    }

    const int rowoff = (lane >> 4) << 3;
#pragma unroll
    for (int j = 0; j < 4; ++j) {
#pragma unroll
        for (int r = 0; r < 8; ++r) {
            int row = m0 + r + rowoff;
            int col = n0 + 16 * j + r16;
            float v = acc[j][r];
            if (bias) v += bias[col];
            C[(size_t)row * ldc + col] = v;
        }
    }
}

// ---------------------------------------------------------------------------
// Kernel 2/4: in-place RMSNorm + 3-axis RoPE on q and k heads of qkv workspace.
// One wave per (token, head, q-or-k). lane handles dims (2*lane, 2*lane+1).
// ---------------------------------------------------------------------------
__global__ __launch_bounds__(256)
void norm_rope(float* __restrict__ qkv, const float* __restrict__ pos,
               const float* __restrict__ qn_w, const float* __restrict__ kn_w) {
    const int lane = threadIdx.x & 31;
    const int wid  = blockIdx.x * 8 + (threadIdx.x >> 5);
    const int t    = wid / (H_ * 2);
    const int rem  = wid % (H_ * 2);
    const int h    = rem >> 1;
    const int isk  = rem & 1;

    float* base = qkv + (size_t)t * LDQKV + isk * DIM + h * DH;
    const float* nw = isk ? kn_w : qn_w;

    const int d0 = lane * 2;
    float x0 = base[d0], x1 = base[d0 + 1];

    float ss = x0 * x0 + x1 * x1;
#pragma unroll
    for (int off = 16; off >= 1; off >>= 1) ss += __shfl_xor(ss, off, 32);

    const float r = rsqrtf(ss * (1.0f / DH) + EPS);
    float y0 = x0 * r * nw[d0];
    float y1 = x1 * r * nw[d0 + 1];

    if (lane < 30) {                        // rope pairs i = lane, dims 2i/2i+1
        const int axis = lane / 10;
        const int f    = lane % 10;
        const float freq = __expf(-(float)f * 0.1f * 9.210340371976184f);
        const float ang  = pos[t * 3 + axis] * freq;
        const float c = cosf(ang), s = sinf(ang);
        const float oe = y0 * c - y1 * s;
        const float oo = y0 * s + y1 * c;
        y0 = oe; y1 = oo;
    }
    base[d0] = y0; base[d0 + 1] = y1;
}

// ---------------------------------------------------------------------------
// Kernel 3/4: fused flash attention. grid = (N/64, H, B); block = 128 (4 waves).
// K/V tiles staged in double-buffered LDS by TDM (wave 0), shared by all 4
// waves. TDM DMA of tile j+1 overlaps the 32 WMMAs of tile j.
// ---------------------------------------------------------------------------
__global__ __launch_bounds__(128)
void flash_attn(const float* __restrict__ qkv, float* __restrict__ out) {
    __shared__ float kt[2][16 * 66];       // padded rows: 64 data + 2 pad
    __shared__ float vt[2][16 * 66];
    __shared__ float pbuf[4][16][17];      // wave-private P transpose buffer

    const int lane   = threadIdx.x & 31;
    const int wv     = threadIdx.x >> 5;
    const int b      = blockIdx.z;
    const int h      = blockIdx.y;
    const int q0     = blockIdx.x * 64 + wv * 16;
    const int r16    = lane & 15;
    const int khalf  = (lane >> 4) << 1;
    const int rowoff = (lane >> 4) << 3;

    const float* Qb = qkv + (size_t)b * N_ * LDQKV + h * DH;
    const float* Kb = Qb + DIM;
    const float* Vb = Qb + 2 * DIM;

    // Q tile (16 x 64) as 16 A-frags, pre-scaled by DH^-1/2
    v2f aQ[16];
#pragma unroll
    for (int t = 0; t < 16; ++t) {
        const float* p = Qb + (size_t)(q0 + r16) * LDQKV + 4 * t + khalf;
        aQ[t].x = p[0] * 0.125f;
        aQ[t].y = p[1] * 0.125f;
    }

    // Prologue: stage first K/V tile into buffer 0.
    if (wv == 0) {
        tdm_load_tile16x64((unsigned)(uintptr_t)&kt[0][0], Kb);
        tdm_load_tile16x64((unsigned)(uintptr_t)&vt[0][0], Vb);
    }

    float mrow[8], lrow[8];
#pragma unroll
    for (int r = 0; r < 8; ++r) { mrow[r] = -INFINITY; lrow[r] = 0.0f; }
    v8f accO[4] = {{}, {}, {}, {}};

    int cur = 0;
    for (int j0 = 0; j0 < N_; j0 += 16) {
        if (wv == 0) __builtin_amdgcn_s_wait_tensorcnt(0);
        __syncthreads();                   // tile `cur` published to all waves
        if (wv == 0 && j0 + 16 < N_) {     // start DMA of next tile now
            tdm_load_tile16x64((unsigned)(uintptr_t)&kt[cur ^ 1][0],
                               Kb + (size_t)(j0 + 16) * LDQKV);
            tdm_load_tile16x64((unsigned)(uintptr_t)&vt[cur ^ 1][0],
                               Vb + (size_t)(j0 + 16) * LDQKV);
        }
        const float* kr = &kt[cur][0];
        const float* vr = &vt[cur][0];

        // S(16x16) = Qs @ K^T, K-frags from LDS (row r16, padded stride 66)
        v8f s = {};
#pragma unroll
        for (int t = 0; t < 16; ++t) {
            v2f bK;
            bK.x = kr[r16 * 66 + 4 * t + khalf];
            bK.y = kr[r16 * 66 + 4 * t + khalf + 1];
            s = wmma4(aQ[t], bK, s);
        }

        // Online softmax: rows of C-frag live across each 16-lane half.
        float fscale[8];
#pragma unroll
        for (int r = 0; r < 8; ++r) {
            float mx = s[r];
#pragma unroll
            for (int off = 8; off >= 1; off >>= 1) mx = fmaxf(mx, __shfl_xor(mx, off, 16));
            const float mnew = fmaxf(mrow[r], mx);
            const float p    = __expf(s[r] - mnew);
            float sum = p;
#pragma unroll
            for (int off = 8; off >= 1; off >>= 1) sum += __shfl_xor(sum, off, 16);
            const float f = __expf(mrow[r] - mnew);
            lrow[r] = lrow[r] * f + sum;
            mrow[r] = mnew;
            fscale[r] = f;
            s[r] = p;
        }
#pragma unroll
        for (int d = 0; d < 4; ++d)
#pragma unroll
            for (int r = 0; r < 8; ++r) accO[d][r] *= fscale[r];

        // Transpose P from C-frag layout to A-frag layout via wave-private LDS
        // (DS ops are in-order within a wave -> no barrier needed).
#pragma unroll
        for (int r = 0; r < 8; ++r) pbuf[wv][r + rowoff][r16] = s[r];
        v2f aP[4];
#pragma unroll
        for (int t = 0; t < 4; ++t) {
            aP[t].x = pbuf[wv][r16][4 * t + khalf];
            aP[t].y = pbuf[wv][r16][4 * t + khalf + 1];
        }

        // O += P @ V, V-frags from LDS (row = key index, cols contiguous)
#pragma unroll
        for (int d = 0; d < 4; ++d) {
#pragma unroll
            for (int t = 0; t < 4; ++t) {
                v2f bV;
                bV.x = vr[(4 * t + khalf) * 66 + d * 16 + r16];
                bV.y = vr[(4 * t + khalf + 1) * 66 + d * 16 + r16];
                accO[d] = wmma4(aP[t], bV, accO[d]);
            }
        }
        cur ^= 1;
    }

    // Normalize by row sums, write (B*N, DIM) attention output
#pragma unroll
    for (int r = 0; r < 8; ++r) {
        const float inv = 1.0f / lrow[r];
        const int grow  = b * N_ + q0 + r + rowoff;
#pragma unroll
        for (int d = 0; d < 4; ++d)
            out[(size_t)grow * DIM + h * DH + d * 16 + r16] = accO[d][r] * inv;
    }
}

// ---------------------------------------------------------------------------
extern "C" void kernel_launch(void* const* d_in, const int* in_sizes, int n_in,
                              void* d_out, int out_size, void* d_ws, size_t ws_size,
                              hipStream_t stream) {
    const float* x      = (const float*)d_in[0];
    const float* pos    = (const float*)d_in[1];
    const float* qkv_w  = (const float*)d_in[2];
    const float* proj_w = (const float*)d_in[3];
    const float* proj_b = (const float*)d_in[4];
    const float* qn_w   = (const float*)d_in[5];
    const float* kn_w   = (const float*)d_in[6];
    float* out = (float*)d_out;

    float* qkv  = (float*)d_ws;                          // (B*N, 2304)
    float* attn = qkv + (size_t)B_ * N_ * LDQKV;         // (B*N, 768)

    // 1) qkv = x @ qkv_w^T
    {
        dim3 grid(LDQKV / 64, (B_ * N_) / 128);
        gemm_wmma<<<grid, 256, 0, stream>>>(x, qkv_w, nullptr, qkv, DIM, LDQKV);
    }
    // 2) rmsnorm(q), rmsnorm(k), 3-axis RoPE in place
    {
        int nwaves = B_ * N_ * H_ * 2;
        norm_rope<<<nwaves / 8, 256, 0, stream>>>(qkv, pos, qn_w, kn_w);
    }
    // 3) fused flash attention -> attn (B*N, 768)
    {
        dim3 grid(N_ / 64, H_, B_);
        flash_attn<<<grid, 128, 0, stream>>>(qkv, attn);
    }
    // 4) out = attn @ proj_w^T + proj_b
    {
        dim3 grid(DIM / 64, (B_ * N_) / 128);
        gemm_wmma<<<grid, 256, 0, stream>>>(attn, proj_w, proj_b, out, DIM, DIM);
    }
}